// RelaxedFacenetLoss_43215960932893
// MI455X (gfx1250) — compile-verified
//
#include <hip/hip_runtime.h>
#include <hip/hip_bf16.h>
#include <math.h>

// Problem constants (match reference)
constexpr int kN = 384;          // batch
constexpr int kD = 128;          // embedding dim
constexpr float kMargin = 0.5f;
constexpr float kTau    = 0.25f;
constexpr float kEps    = 1e-12f;

// Workspace layout (in floats)
constexpr int kXN   = 0;                    // xn_src [N*D], xn_emb [N*D]
constexpr int kDM   = 2 * kN * kD;          // d_src [N*N], d_emb [N*N]  (1 - sim)
constexpr int kPW   = kDM + 2 * kN * kN;    // pw [N*N]
constexpr int kSC   = kPW + kN * kN;        // scalars (16 floats)
constexpr int kPART = kSC + 16;             // per-row partial sums [N]

typedef __attribute__((ext_vector_type(2))) float v2f;
typedef __attribute__((ext_vector_type(8))) float v8f;

// ---------------------------------------------------------------------------
// 1) Row L2-normalize both inputs.  grid = (N, 2), block = D (=128, 4 waves)
// ---------------------------------------------------------------------------
__global__ void k_normalize(const float* __restrict__ src,
                            const float* __restrict__ emb,
                            float* __restrict__ ws) {
    const int row   = blockIdx.x;
    const int which = blockIdx.y;                  // 0 = source, 1 = student
    const float* x  = (which == 0) ? src : emb;
    float* xn       = ws + kXN + which * (kN * kD);

    __shared__ float red[kD];
    const int t = threadIdx.x;
    const float v = x[row * kD + t];
    red[t] = v * v;
    __syncthreads();
    for (int s = kD / 2; s > 0; s >>= 1) {
        if (t < s) red[t] += red[t + s];
        __syncthreads();
    }
    const float scale = 1.0f / fmaxf(sqrtf(red[0]), kEps);
    xn[row * kD + t] = v * scale;
}

// ---------------------------------------------------------------------------
// 2) Gram matrix via fp32 WMMA, stored as distance d = 1 - xn·xnT.
//    grid = (N/16, N/16, 2), block = 32 (one wave -> one 16x16 tile).
//    K stepped by 4 -> 32 chained V_WMMA_F32_16X16X4_F32 per tile.
// ---------------------------------------------------------------------------
__global__ void k_dist_wmma(const float* __restrict__ ws_ro,
                            float* __restrict__ ws) {
    const float* xn = ws_ro + kXN + blockIdx.z * (kN * kD);
    float* dmat     = ws + kDM + blockIdx.z * (kN * kN);

    const int lane = threadIdx.x;        // 0..31
    const int half = lane >> 4;          // 0 | 1
    const int m    = lane & 15;

    // A fragment rows come from the row tile, B fragment "columns" from the
    // column tile of the same matrix (B = Xn^T tile => B[k,n] = Xn[col+n, k]).
    const float* pa = xn + (blockIdx.y * 16 + m) * kD + 2 * half;
    const float* pb = xn + (blockIdx.x * 16 + m) * kD + 2 * half;

    v8f c = {0.f, 0.f, 0.f, 0.f, 0.f, 0.f, 0.f, 0.f};

#pragma unroll 8
    for (int kk = 0; kk < kD; kk += 4) {
        v2f a, b;
        a.x = pa[kk];     a.y = pa[kk + 1];   // K = kk+2*half, kk+2*half+1
        b.x = pb[kk];     b.y = pb[kk + 1];
        // 8 args: (neg_a, A, neg_b, B, c_mod, C, reuse_a, reuse_b)
        c = __builtin_amdgcn_wmma_f32_16x16x4_f32(
                false, a, false, b, (short)0, c, false, false);
    }

    const int col = blockIdx.x * 16 + m;
#pragma unroll
    for (int v = 0; v < 8; ++v) {
        const int row = blockIdx.y * 16 + v + 8 * half;   // C/D VGPR layout
        dmat[row * kN + col] = 1.0f - c[v];
    }
}

// ---------------------------------------------------------------------------
// 3) mean(d_src) -> store s = 1 / (max(mean, eps) * tau).  Single block,
//    fixed-order strided accumulation + tree reduction (deterministic).
// ---------------------------------------------------------------------------
__global__ void k_mean_scale(float* __restrict__ ws) {
    const float* dsrc = ws + kDM;
    __shared__ float red[256];
    const int t = threadIdx.x;
    float acc = 0.0f;
    for (int i = t; i < kN * kN; i += 256) acc += dsrc[i];
    red[t] = acc;
    __syncthreads();
    for (int s = 128; s > 0; s >>= 1) {
        if (t < s) red[t] += red[t + s];
        __syncthreads();
    }
    if (t == 0) {
        const float mean = red[0] / (float)(kN * kN);
        ws[kSC] = 1.0f / (fmaxf(mean, kEps) * kTau);
    }
}

// ---------------------------------------------------------------------------
// 4) pw[i,j] = exp(-d_src[i,j] * s)
// ---------------------------------------------------------------------------
__global__ void k_pw(float* __restrict__ ws) {
    const float s     = ws[kSC];
    const float* dsrc = ws + kDM;
    float* pw         = ws + kPW;
    const int idx = blockIdx.x * blockDim.x + threadIdx.x;
    if (idx < kN * kN) pw[idx] = expf(-dsrc[idx] * s);
}

// ---------------------------------------------------------------------------
// 5) The N^3 kernel.  One block per row i (384 threads = 12 waves).
//    d_emb[i,:] and npw[i,:] live in LDS; thread j scans all k:
//      tl = margin - d[i,k] + d[i,j];  valid iff 0 < tl <= margin
//      m[j] = max_k(tl * npw[i,k] * valid)
//    contribution = pw[i,j]^2 * m[j]   (pw factored out of the max; >= 0)
// ---------------------------------------------------------------------------
__global__ void k_rowmax(float* __restrict__ ws) {
    const int i = blockIdx.x;
    const float* demb  = ws + kDM + kN * kN + i * kN;
    const float* pwrow = ws + kPW + i * kN;

    __shared__ float sd[kN];
    __shared__ float sn[kN];
    __shared__ float red[kN];

    const int t = threadIdx.x;           // j = t, 0..383
    sd[t] = demb[t];
    sn[t] = 1.0f - pwrow[t];             // neg pair weight
    __syncthreads();

    const float dj = sd[t];
    float mx = 0.0f;
#pragma unroll 8
    for (int k = 0; k < kN; ++k) {
        const float x   = kMargin - sd[k] + dj;          // triplet loss pre-relu
        const float val = (x > 0.0f && x <= kMargin) ? x * sn[k] : 0.0f;
        mx = fmaxf(mx, val);
    }
    const float p = pwrow[t];
    red[t] = p * p * mx;                 // ppsl[i,j] * pw[i,j]
    __syncthreads();

    if (t < 128) red[t] += red[t + 128] + red[t + 256];  // 384 -> 128
    __syncthreads();
    for (int s = 64; s > 0; s >>= 1) {
        if (t < s) red[t] += red[t + s];
        __syncthreads();
    }
    if (t == 0) ws[kPART + i] = red[0];
}

// ---------------------------------------------------------------------------
// 6) Final: loss = safe_div( sum(partials), sum(pw) ).  Single block.
// ---------------------------------------------------------------------------
__global__ void k_final(const float* __restrict__ ws, float* __restrict__ out) {
    __shared__ float rn[256];
    __shared__ float rd[256];
    const int t = threadIdx.x;
    float num = 0.0f, den = 0.0f;
    for (int i = t; i < kN; i += 256)       num += ws[kPART + i];
    const float* pw = ws + kPW;
    for (int i = t; i < kN * kN; i += 256)  den += pw[i];
    rn[t] = num; rd[t] = den;
    __syncthreads();
    for (int s = 128; s > 0; s >>= 1) {
        if (t < s) { rn[t] += rn[t + s]; rd[t] += rd[t + s]; }
        __syncthreads();
    }
    if (t == 0) out[0] = (rd[0] > 0.0f) ? (rn[0] / rd[0]) : 0.0f;
}

// ---------------------------------------------------------------------------
extern "C" void kernel_launch(void* const* d_in, const int* in_sizes, int n_in,
                              void* d_out, int out_size, void* d_ws, size_t ws_size,
                              hipStream_t stream) {
    const float* src = (const float*)d_in[0];   // source_embeddings [384,128] f32
    const float* emb = (const float*)d_in[1];   // embeddings        [384,128] f32
    float* out = (float*)d_out;                 // scalar loss
    float* ws  = (float*)d_ws;

    (void)in_sizes; (void)n_in; (void)out_size; (void)ws_size;

    // 1) normalize rows of both matrices
    k_normalize<<<dim3(kN, 2, 1), kD, 0, stream>>>(src, emb, ws);
    // 2) fp32 WMMA Gram -> distances (z: 0=src, 1=emb)
    k_dist_wmma<<<dim3(kN / 16, kN / 16, 2), 32, 0, stream>>>(ws, ws);
    // 3) mean of teacher distances -> scale scalar
    k_mean_scale<<<1, 256, 0, stream>>>(ws);
    // 4) teacher pair weights
    k_pw<<<(kN * kN) / 256, 256, 0, stream>>>(ws);
    // 5) per-row semihard max scan (the N^3 part)
    k_rowmax<<<kN, kN, 0, stream>>>(ws);
    // 6) weighted sum / safe division -> loss
    k_final<<<1, 256, 0, stream>>>(ws, out);
}